// FakeQuantRTN_29592324670161
// MI455X (gfx1250) — compile-verified
//
#include <hip/hip_runtime.h>
#include <math.h>

// FakeQuantRTN (AWQ-style MSE grid search), w: (4096, 11008) fp32.
// GROUP_SIZE=128, 4-bit symmetric (MAX_INT=7), 80 shrink ratios, |.|^2.4 norm.
//
// Mapping: 8 lanes/group, 16 elems/lane, 4 groups/wave, 8 waves/block
// -> 32 groups (16KB) per 256-thread block, 11008 blocks total.
// Data path: CDNA5 async global->LDS staging (ASYNCcnt), then registers for
// the entire 80-iteration search (w read from HBM exactly once).
// Compute-bound on the transcendental pipe: |d|^2.4 = v_exp(2.4 * v_log(|d|)).

#define TPB            256
#define GSIZE          128
#define GRPS_PER_WAVE  4
#define GRPS_PER_BLOCK 32
#define QMAX           7.0f
#define QEPS           1e-10f
#define NSTEPS         80

// Hardware transcendentals: v_log_f32 computes log2(x); v_exp_f32 computes 2^x.
__device__ __forceinline__ float fast_log2(float x) {
#if __has_builtin(__builtin_amdgcn_logf)
  return __builtin_amdgcn_logf(x);
#else
  return log2f(x);
#endif
}
__device__ __forceinline__ float fast_exp2(float x) {
#if __has_builtin(__builtin_amdgcn_exp2f)
  return __builtin_amdgcn_exp2f(x);
#else
  return exp2f(x);
#endif
}
// v_rcp_f32: ~1 ulp; only feeds the pre-round product, full div kept for scale.
__device__ __forceinline__ float fast_rcp(float x) {
#if __has_builtin(__builtin_amdgcn_rcpf)
  return __builtin_amdgcn_rcpf(x);
#else
  return 1.0f / x;
#endif
}
// Single v_med3_f32 clamp (operands always finite here).
__device__ __forceinline__ float clamp7(float q) {
#if __has_builtin(__builtin_amdgcn_fmed3f)
  return __builtin_amdgcn_fmed3f(q, -QMAX, QMAX);
#else
  return fminf(fmaxf(q, -QMAX), QMAX);
#endif
}

__device__ __forceinline__ float sub8_max(float v) {
  v = fmaxf(v, __shfl_xor(v, 1, 32));
  v = fmaxf(v, __shfl_xor(v, 2, 32));
  v = fmaxf(v, __shfl_xor(v, 4, 32));
  return v;
}

__device__ __forceinline__ float sub8_sum(float v) {
  v += __shfl_xor(v, 1, 32);
  v += __shfl_xor(v, 2, 32);
  v += __shfl_xor(v, 4, 32);
  return v;
}

// round-half-even, clamp to [-7,7], dequant, STE arithmetic (w + (wq - w)).
// True division to mirror the reference's final quant_dequant exactly.
__device__ __forceinline__ float qdq_ste(float wv, float s) {
  float q = clamp7(rintf(wv / s));       // v_rndne_f32 (matches jnp.round)
  float wq = q * s;
  return wv + (wq - wv);
}

__global__ __launch_bounds__(TPB, 2)
void fakequant_rtn_kernel(const float* __restrict__ w, float* __restrict__ out) {
  __shared__ float smem[GRPS_PER_BLOCK * GSIZE];   // 16 KB staging buffer

  const int tid  = threadIdx.x;
  const int wid  = tid >> 5;
  const int lane = tid & 31;
  const long long blockBase = (long long)blockIdx.x * (GRPS_PER_BLOCK * GSIZE);
  const int waveBase = wid * (GRPS_PER_WAVE * GSIZE);   // float offset in smem

  // ---- Stage this wave's 2 KB: async global -> LDS (tracked by ASYNCcnt) ----
  {
    // AS3 byte offset = low 32 bits of the generic (flat) LDS pointer.
    unsigned ldsBase = (unsigned)(size_t)(&smem[waveBase]);
    unsigned long long gBase =
        (unsigned long long)(size_t)(w + blockBase + waveBase);
#pragma unroll
    for (int i = 0; i < 4; ++i) {
      unsigned byteOff = (unsigned)(lane * 16 + i * 512);
      unsigned ldsOff = ldsBase + byteOff;
      unsigned long long gaddr = gBase + byteOff;
      asm volatile("global_load_async_to_lds_b128 %0, %1, off"
                   :: "v"(ldsOff), "v"(gaddr)
                   : "memory");
    }
    // Wave only reads LDS it staged itself -> per-wave wait, no barrier.
    asm volatile("s_wait_asynccnt 0x0" ::: "memory");
  }

  // ---- Unpack my 16 elements (group s, lane j owns float4 chunks j,j+8,j+16,j+24)
  const int s = lane >> 3;          // group within wave
  const int j = lane & 7;           // lane within 8-lane subgroup
  const int grpF = waveBase + s * GSIZE;

  float x[16];
#pragma unroll
  for (int i = 0; i < 4; ++i) {
    float4 v = *reinterpret_cast<const float4*>(&smem[grpF + (j + 8 * i) * 4]);
    x[4 * i + 0] = v.x;
    x[4 * i + 1] = v.y;
    x[4 * i + 2] = v.z;
    x[4 * i + 3] = v.w;
  }

  // ---- Group amax (3-step butterfly within the 8-lane subgroup) ----
  float amax = 0.0f;
#pragma unroll
  for (int e = 0; e < 16; ++e) amax = fmaxf(amax, fabsf(x[e]));
  amax = sub8_max(amax);
  amax = fmaxf(amax, QEPS);

  // ---- 80-ratio MSE grid search (transcendental-bound inner loop) ----
  float bestErr   = INFINITY;               // first ratio always wins vs inf
  float bestScale = amax / QMAX;
  for (int r = 0; r < NSTEPS; ++r) {
    float ratio = 1.0f - ((float)r / (float)NSTEPS) * 0.8f;
    float scale = fmaxf(amax * ratio / QMAX, QEPS);  // exact div: matches reference scale
    float inv   = fast_rcp(scale);                   // v_rcp_f32, feeds rounding only
    float err   = 0.0f;
#pragma unroll
    for (int e = 0; e < 16; ++e) {
      float q = clamp7(rintf(x[e] * inv));           // v_rndne + v_med3
      float d = fabsf(fmaf(q, scale, -x[e]));        // |wq - w| (abs folds into v_log modifier)
      // |d|^2.4 = exp2(2.4*log2 d); d==0 -> log2=-inf -> exp2=+0 (correct)
      err += fast_exp2(2.4f * fast_log2(d));         // v_log_f32 / v_exp_f32
    }
    err = sub8_sum(err);                             // 3 shuffles, all lanes agree
    if (err < bestErr) { bestErr = err; bestScale = scale; }  // strict <: earliest ratio wins ties
  }

  // ---- Final quant-dequant with best scale, coalesced float4 stores ----
#pragma unroll
  for (int i = 0; i < 4; ++i) {
    float4 o;
    o.x = qdq_ste(x[4 * i + 0], bestScale);
    o.y = qdq_ste(x[4 * i + 1], bestScale);
    o.z = qdq_ste(x[4 * i + 2], bestScale);
    o.w = qdq_ste(x[4 * i + 3], bestScale);
    *reinterpret_cast<float4*>(&out[blockBase + grpF + (j + 8 * i) * 4]) = o;
  }
}

extern "C" void kernel_launch(void* const* d_in, const int* in_sizes, int n_in,
                              void* d_out, int out_size, void* d_ws, size_t ws_size,
                              hipStream_t stream) {
  const float* w = (const float*)d_in[0];
  float* out     = (float*)d_out;

  const int n        = in_sizes[0];               // 4096 * 11008 = 45,088,768
  const int nGroups  = n / GSIZE;                 // 352,256
  const int nBlocks  = nGroups / GRPS_PER_BLOCK;  // 11,008 (exact)

  fakequant_rtn_kernel<<<nBlocks, TPB, 0, stream>>>(w, out);
}